// DenseFlashAttention_57492432224942
// MI455X (gfx1250) — compile-verified
//
#include <hip/hip_runtime.h>
#include <math.h>

// ---------------------------------------------------------------------------
// Graph attention (segment softmax over incoming edges) for MI455X / gfx1250.
//   N = 100000 nodes, E = 1600000 edges, D = 64, SCALE = 1/8.
// Dense GEMMs via V_WMMA_F32_16X16X4_F32 (fp32 A/B/C/D, K=4 per instruction).
// Edge phase: wave-per-edge gathers (L2-resident: working set ~130MB < 192MB L2)
// with fp32 global atomics for the segment reductions.
// ---------------------------------------------------------------------------

#define DIM 64
#define SCALE 0.125f  // 64^-0.5

typedef float v2f __attribute__((ext_vector_type(2)));
typedef float v8f __attribute__((ext_vector_type(8)));

// ---- order-preserving float <-> uint mapping for atomicMax on floats -------
__device__ __forceinline__ unsigned f2key(float f) {
    int i = __float_as_int(f);
    return (unsigned)(i ^ ((i >> 31) | 0x80000000));
}
__device__ __forceinline__ float key2f(unsigned u) {
    int i = (u & 0x80000000u) ? (int)(u ^ 0x80000000u) : ~(int)u;
    return __int_as_float(i);
}
#define KEY_NEG_INF 0x007FFFFFu  // f2key(-inf)

// ---------------------------------------------------------------------------
// 16x16 output tile of  A[16x64] * B[64x16]  using 16 chained WMMA f32 steps.
// Layouts per CDNA5 ISA 7.12.2:
//   A 16x4 : lane(0-15)=row M, v0/v1 = K{0,1}; lane(16-31): K{2,3}
//   B 4x16 : v0 lanes0-15 = row K0, lanes16-31 = row K2; v1 = K1/K3
//   C/D    : vgpr j, lane L -> (M = j + 8*(L>>4), N = L&15)
// ---------------------------------------------------------------------------
__device__ __forceinline__ v8f wmma_a64_b16(const float* __restrict__ A,  // 16 x 64, row-major, ld=64
                                            const float* __restrict__ B,  // 64 x 16 slice of 64x64, ld=64
                                            int half, int idx) {
    v8f acc = {};
#pragma unroll
    for (int k = 0; k < DIM; k += 4) {
        const int ka = k + 2 * half;
        v2f a = *(const v2f*)(A + idx * DIM + ka);     // A[idx][ka..ka+1]
        v2f b;
        b.x = B[ka * DIM + idx];                        // B[ka][idx]
        b.y = B[(ka + 1) * DIM + idx];                  // B[ka+1][idx]
        acc = __builtin_amdgcn_wmma_f32_16x16x4_f32(
            /*neg_a=*/false, a, /*neg_b=*/false, b,
            /*c_mod=*/(short)0, acc, /*reuse_a=*/false, /*reuse_b=*/false);
    }
    return acc;
}

// ---------------------------------------------------------------------------
// Kernel 1: zero accumulators, init segment-max keys.
// ---------------------------------------------------------------------------
__global__ void ga_init_kernel(float* __restrict__ num, float* __restrict__ denom,
                               unsigned* __restrict__ m, int N) {
    const int tid = blockIdx.x * blockDim.x + threadIdx.x;
    const int stride = gridDim.x * blockDim.x;
    const int total = N * DIM;
    for (int t = tid; t < total; t += stride) num[t] = 0.0f;
    for (int t = tid; t < N; t += stride) { denom[t] = 0.0f; m[t] = KEY_NEG_INF; }
}

// ---------------------------------------------------------------------------
// Kernel 2: Q = x@Wq, K = x@Wk, V = x@Wv.  128 threads = 4 waves; wave w owns
// column tile w of the 16-row node tile.  (N % 16 == 0 for this problem.)
// ---------------------------------------------------------------------------
__global__ void ga_qkv_kernel(const float* __restrict__ x,
                              const float* __restrict__ Wq,
                              const float* __restrict__ Wk,
                              const float* __restrict__ Wv,
                              float* __restrict__ Q, float* __restrict__ K,
                              float* __restrict__ V) {
    const int row0 = blockIdx.x * 16;
    const int wave = threadIdx.x >> 5;     // 0..3 -> column tile
    const int lane = threadIdx.x & 31;
    const int half = lane >> 4;
    const int idx  = lane & 15;
    const int col0 = wave * 16;
    const float* A = x + (size_t)row0 * DIM;

    v8f cq = wmma_a64_b16(A, Wq + col0, half, idx);
    v8f ck = wmma_a64_b16(A, Wk + col0, half, idx);
    v8f cv = wmma_a64_b16(A, Wv + col0, half, idx);

#pragma unroll
    for (int j = 0; j < 8; ++j) {
        const size_t o = (size_t)(row0 + j + 8 * half) * DIM + col0 + idx;
        Q[o] = cq[j];
        K[o] = ck[j];
        V[o] = cv[j];
    }
}

// ---------------------------------------------------------------------------
// Kernel 3: per-edge logits + segment max.  One wave per edge (grid-stride):
// lanes split the 64-dim dot product (2 floats/lane, coalesced 128B loads),
// butterfly reduce, lane 0 stores s[e] and atomicMax's the receiver's max.
// ---------------------------------------------------------------------------
__global__ void ga_logits_kernel(const int* __restrict__ ei,  // [2, E]
                                 const float* __restrict__ Q,
                                 const float* __restrict__ K,
                                 float* __restrict__ s, unsigned* __restrict__ m,
                                 int E) {
    const int lane = threadIdx.x & 31;
    const int gw = (blockIdx.x * blockDim.x + threadIdx.x) >> 5;
    const int nw = (gridDim.x * blockDim.x) >> 5;
    for (int e = gw; e < E; e += nw) {
        const int snd = ei[e];
        const int rcv = ei[E + e];
        const float* q = Q + (size_t)rcv * DIM;
        const float* k = K + (size_t)snd * DIM;
        float p = q[lane] * k[lane] + q[lane + 32] * k[lane + 32];
#pragma unroll
        for (int off = 16; off > 0; off >>= 1) p += __shfl_xor(p, off, 32);
        p *= SCALE;
        if (lane == 0) {
            s[e] = p;
            atomicMax(m + rcv, f2key(p));
        }
    }
}

// ---------------------------------------------------------------------------
// Kernel 4: p = exp(s - m[recv]); denom[recv] += p; num[recv] += p * V[send].
// One wave per edge; 2 fp32 atomic adds per lane into the receiver row.
// ---------------------------------------------------------------------------
__global__ void ga_accum_kernel(const int* __restrict__ ei,
                                const float* __restrict__ V,
                                const float* __restrict__ s,
                                const unsigned* __restrict__ m,
                                float* __restrict__ num, float* __restrict__ denom,
                                int E) {
    const int lane = threadIdx.x & 31;
    const int gw = (blockIdx.x * blockDim.x + threadIdx.x) >> 5;
    const int nw = (gridDim.x * blockDim.x) >> 5;
    for (int e = gw; e < E; e += nw) {
        const int snd = ei[e];
        const int rcv = ei[E + e];
        // rcv has at least this edge, so m[rcv] is finite.
        const float p = __expf(s[e] - key2f(m[rcv]));
        if (lane == 0) atomicAdd(denom + rcv, p);
        const float* v = V + (size_t)snd * DIM;
        float* n = num + (size_t)rcv * DIM;
        atomicAdd(n + lane, p * v[lane]);
        atomicAdd(n + lane + 32, p * v[lane + 32]);
    }
}

// ---------------------------------------------------------------------------
// Kernel 5: H = num/denom (0 where denom==0);  out = x + H @ Wo   (fused WMMA)
// ---------------------------------------------------------------------------
__global__ void ga_proj_kernel(const float* __restrict__ x,
                               const float* __restrict__ num,
                               const float* __restrict__ denom,
                               const float* __restrict__ Wo,
                               float* __restrict__ out) {
    const int row0 = blockIdx.x * 16;
    const int wave = threadIdx.x >> 5;
    const int lane = threadIdx.x & 31;
    const int half = lane >> 4;
    const int idx  = lane & 15;
    const int col0 = wave * 16;

    // Per-lane A-row scaling factor (row = row0 + idx for A operand).
    const int arow = row0 + idx;
    const float d = denom[arow];
    const float inv = (d > 0.0f) ? (1.0f / fmaxf(d, 1e-30f)) : 0.0f;
    const float* nrow = num + (size_t)arow * DIM;

    v8f acc = {};
#pragma unroll
    for (int k = 0; k < DIM; k += 4) {
        const int ka = k + 2 * half;
        v2f a;
        a.x = nrow[ka] * inv;
        a.y = nrow[ka + 1] * inv;
        v2f b;
        b.x = Wo[ka * DIM + col0 + idx];
        b.y = Wo[(ka + 1) * DIM + col0 + idx];
        acc = __builtin_amdgcn_wmma_f32_16x16x4_f32(false, a, false, b,
                                                    (short)0, acc, false, false);
    }
#pragma unroll
    for (int j = 0; j < 8; ++j) {
        const size_t o = (size_t)(row0 + j + 8 * half) * DIM + col0 + idx;
        out[o] = x[o] + acc[j];
    }
}

// ---------------------------------------------------------------------------
extern "C" void kernel_launch(void* const* d_in, const int* in_sizes, int n_in,
                              void* d_out, int out_size, void* d_ws, size_t ws_size,
                              hipStream_t stream) {
    const float* x  = (const float*)d_in[0];
    const int*   ei = (const int*)d_in[1];
    const float* Wq = (const float*)d_in[2];
    const float* Wk = (const float*)d_in[3];
    const float* Wv = (const float*)d_in[4];
    const float* Wo = (const float*)d_in[5];
    float* out = (float*)d_out;

    const int N = in_sizes[0] / DIM;  // 100000
    const int E = in_sizes[1] / 2;    // 1600000

    // Workspace layout (floats): Q | K | V | num | denom | m | s
    float* Q     = (float*)d_ws;
    float* K     = Q + (size_t)N * DIM;
    float* V     = K + (size_t)N * DIM;
    float* num   = V + (size_t)N * DIM;
    float* denom = num + (size_t)N * DIM;
    unsigned* m  = (unsigned*)(denom + N);
    float* s     = (float*)(m + N);

    ga_init_kernel<<<1024, 256, 0, stream>>>(num, denom, m, N);
    ga_qkv_kernel<<<N / 16, 128, 0, stream>>>(x, Wq, Wk, Wv, Q, K, V);
    ga_logits_kernel<<<2048, 256, 0, stream>>>(ei, Q, K, s, m, E);
    ga_accum_kernel<<<2048, 256, 0, stream>>>(ei, V, s, m, num, denom, E);
    ga_proj_kernel<<<N / 16, 128, 0, stream>>>(x, num, denom, Wo, out);
}